// YalmSelfAttention_30571577213836
// MI455X (gfx1250) — compile-verified
//
#include <hip/hip_runtime.h>
#include <math.h>

// ---------------------------------------------------------------- types
typedef float v2f __attribute__((ext_vector_type(2)));
typedef float v8f __attribute__((ext_vector_type(8)));
typedef unsigned int u32x4 __attribute__((ext_vector_type(4)));
typedef int i32x4 __attribute__((ext_vector_type(4)));
typedef int i32x8 __attribute__((ext_vector_type(8)));

// ---------------------------------------------------------------- config
#define S_LEN   2048
#define BATCH   2
#define HID     2048
#define NHEAD   16
#define HN      128
#define M_ROWS  (S_LEN * BATCH)   // 4096
#define QKV_N   (3 * HID)         // 6144
#define ATT_SCALE 0.08838834764831845f  // COEFF/NORM_FACTOR = 1/sqrt(HN)
#define MASK_FILL (-10000.0f)

// GEMM tiling: block 64x128, 8 waves in 2x4 grid, each wave 32x32 (2x2 WMMA tiles)
#define BM 64
#define BN 128
#define KT 32
#define KPAD (KT + 2)

// Attention tiling
#define QROWS 128   // query rows per block (8 waves x 16)
#define KV 16       // kv tile

static __device__ __forceinline__ v8f wmma_f32(v2f a, v2f b, v8f c) {
  // D = A(16x4) * B(4x16) + C(16x16), fp32 matrix pipe (CDNA5)
  return __builtin_amdgcn_wmma_f32_16x16x4_f32(
      /*neg_a=*/false, a, /*neg_b=*/false, b,
      /*c_mod=*/(short)0, c, /*reuse_a=*/false, /*reuse_b=*/false);
}

// ---------------------------------------------------------------- TDM helper
// 2D tile load via Tensor Data Mover: 16 x 128 fp32 tile out of a
// (2048 x 128) row-major tensor, with 2-DWORD LDS row padding (-> HN+2 rows).
// D# layout per CDNA5 ISA.
// This toolchain exposes the 6-arg builtin:
//   (u32x4 g0, i32x8 g1, i32x4 g2, i32x4 g3, i32x8 g4, i32 cpol)
static __device__ __forceinline__ void tdm_load_tile_128(unsigned lds_off,
                                                         const float* gptr) {
  unsigned long long ga = (unsigned long long)(size_t)gptr;
  u32x4 g0;
  g0[0] = 1u;                                   // count=1, user descriptor
  g0[1] = lds_off;                              // lds_addr (bytes)
  g0[2] = (unsigned)ga;                         // global_addr[31:0]
  g0[3] = (unsigned)((ga >> 32) & 0x01FFFFFFu)  // global_addr[56:32]
          | 0x80000000u;                        // type=2 ("image")
  i32x8 g1;
  g1[0] = (int)(0x20000u        // data_size=2 -> 4 bytes
                | (1u << 20)    // pad_enable
                | (6u << 22)    // pad_interval: every 128 DWORDs
                | (1u << 25));  // pad_amount: 2 DWORDs
  g1[1] = (int)(128u << 16);    // tensor_dim0 = 128 [79:48]
  g1[2] = (int)(2048u << 16);   // tensor_dim1 = 2048 [111:80]
  g1[3] = (int)(128u << 16);    // tile_dim0 = 128 [127:112]
  g1[4] = (int)KV;              // tile_dim1 = 16 [143:128]
  g1[5] = 128;                  // tensor_dim0_stride [207:160]
  g1[6] = 0;
  g1[7] = 0;
  i32x4 z4 = {};
  i32x8 z8 = {};
  __builtin_amdgcn_tensor_load_to_lds(g0, g1, z4, z4, z8, 0);
}

// ================================================================ kernel 1
// mixed = hidden(M_ROWS x HID) * w_qkv(QKV_N x HID)^T + b_qkv,
// scattered into q/k/v workspaces laid out [b][head][s][hn].
__global__ __launch_bounds__(256)
void qkv_gemm_kernel(const float* __restrict__ A,
                     const float* __restrict__ W,
                     const float* __restrict__ bias,
                     float* __restrict__ qws,
                     float* __restrict__ kws,
                     float* __restrict__ vws)
{
  __shared__ float As[BM][KPAD];
  __shared__ float Bs[BN][KPAD];

  const int rowBase = blockIdx.y * BM;
  const int colBase = blockIdx.x * BN;
  const int t    = threadIdx.x;
  const int wave = t >> 5;        // 0..7
  const int lane = t & 31;
  const int wm   = wave >> 2;     // 0..1  (32-row strip)
  const int wn   = wave & 3;      // 0..3  (32-col strip)
  const int grp  = lane >> 4;     // 0..1
  const int ln16 = lane & 15;

  v8f acc[2][2];
#pragma unroll
  for (int sm = 0; sm < 2; ++sm)
#pragma unroll
    for (int sn = 0; sn < 2; ++sn)
#pragma unroll
      for (int i = 0; i < 8; ++i) acc[sm][sn][i] = 0.0f;

  for (int k0 = 0; k0 < HID; k0 += KT) {
    __syncthreads();
    for (int idx = t; idx < BM * KT; idx += 256) {
      int r = idx >> 5, c = idx & 31;
      As[r][c] = A[(size_t)(rowBase + r) * HID + k0 + c];
      __builtin_prefetch(&A[(size_t)(rowBase + r) * HID + k0 + KT + c], 0, 0);
    }
    for (int idx = t; idx < BN * KT; idx += 256) {
      int r = idx >> 5, c = idx & 31;
      Bs[r][c] = W[(size_t)(colBase + r) * HID + k0 + c];
      __builtin_prefetch(&W[(size_t)(colBase + r) * HID + k0 + KT + c], 0, 0);
    }
    __syncthreads();

#pragma unroll
    for (int kk = 0; kk < KT; kk += 4) {
      const int kc = kk + 2 * grp;
      v2f a0, a1, b0, b1;
      a0.x = As[wm * 32 + ln16][kc];      a0.y = As[wm * 32 + ln16][kc + 1];
      a1.x = As[wm * 32 + 16 + ln16][kc]; a1.y = As[wm * 32 + 16 + ln16][kc + 1];
      b0.x = Bs[wn * 32 + ln16][kc];      b0.y = Bs[wn * 32 + ln16][kc + 1];
      b1.x = Bs[wn * 32 + 16 + ln16][kc]; b1.y = Bs[wn * 32 + 16 + ln16][kc + 1];
      acc[0][0] = wmma_f32(a0, b0, acc[0][0]);
      acc[0][1] = wmma_f32(a0, b1, acc[0][1]);
      acc[1][0] = wmma_f32(a1, b0, acc[1][0]);
      acc[1][1] = wmma_f32(a1, b1, acc[1][1]);
    }
  }

#pragma unroll
  for (int sm = 0; sm < 2; ++sm) {
#pragma unroll
    for (int sn = 0; sn < 2; ++sn) {
#pragma unroll
      for (int i = 0; i < 8; ++i) {
        int m = i + 8 * grp;
        int r = rowBase + wm * 32 + sm * 16 + m;     // = s*BATCH + b
        int o = colBase + wn * 32 + sn * 16 + ln16;  // 0..6143
        float val = acc[sm][sn][i] + bias[o];
        int s    = r >> 1;
        int bidx = r & 1;
        int nh    = o / (3 * HN);
        int j     = o % (3 * HN);
        int which = j / HN;
        int hn    = j % HN;
        size_t dst = (((size_t)bidx * NHEAD + nh) * S_LEN + s) * HN + hn;
        if (which == 0)      qws[dst] = val;
        else if (which == 1) kws[dst] = val;
        else                 vws[dst] = val;
      }
    }
  }
}

// ================================================================ kernel 2
// Rotary embedding, in place on q and k. One thread per (head, s, pair i).
__global__ __launch_bounds__(256)
void rope_kernel(float* __restrict__ qws, float* __restrict__ kws)
{
  size_t idx = (size_t)blockIdx.x * 256 + threadIdx.x;
  const size_t total = (size_t)BATCH * NHEAD * S_LEN * (HN / 2);
  if (idx >= total) return;
  int    i    = (int)(idx & 63);            // pair index 0..63
  size_t rest = idx >> 6;
  int    s    = (int)(rest % S_LEN);
  size_t head = rest / S_LEN;

  float freq = __powf(1e-4f, (2.0f * i) / (float)HN);
  float ang  = (float)s * freq;
  float c = __cosf(ang), sn = __sinf(ang);

  size_t base = (head * S_LEN + s) * HN;
  float ql = qws[base + i], qr = qws[base + i + HN / 2];
  qws[base + i]          = c * ql - sn * qr;
  qws[base + i + HN / 2] = sn * ql + c * qr;
  float kl = kws[base + i], kr = kws[base + i + HN / 2];
  kws[base + i]          = c * kl - sn * kr;
  kws[base + i + HN / 2] = sn * kl + c * kr;
}

// ================================================================ kernel 3
// Causal flash attention per (b, head, 128-query-row tile).
// Q staged once in LDS; K/V tiles double-buffered through the Tensor Data
// Mover: wave 0 issues tile t+1 before waiting on tile t, so the TDM DMA
// overlaps the WMMA work of the current tile (TENSORcnt is per-wave,
// in-order: wait <= 2 means the older pair has landed).
__global__ __launch_bounds__(256)
void attn_kernel(const float* __restrict__ qws,
                 const float* __restrict__ kws,
                 const float* __restrict__ vws,
                 float* __restrict__ ctx)
{
  __shared__ float Qs[QROWS][HN + 2];
  __shared__ float Ks[2][KV][HN + 2];
  __shared__ float Vs[2][KV][HN + 2];
  __shared__ float Ssc[8][16][17];
  __shared__ float aLds[8][16];
  __shared__ float lLds[8][16];

  const int t    = threadIdx.x;
  const int wave = t >> 5;
  const int lane = t & 31;
  const int grp  = lane >> 4;
  const int ln16 = lane & 15;

  const int head = blockIdx.z * NHEAD + blockIdx.y;
  const size_t hbase = (size_t)head * S_LEN * HN;
  const int qBlock = blockIdx.x * QROWS;
  const int qWave  = qBlock + wave * 16;

  const float* Qp = qws + hbase;
  const float* Kp = kws + hbase;
  const float* Vp = vws + hbase;

  // ---- stage this block's 128 Q rows once
  for (int idx = t; idx < QROWS * HN; idx += 256) {
    int r = idx >> 7, c = idx & 127;
    Qs[r][c] = Qp[(size_t)(qBlock + r) * HN + c];
  }

  v8f ctxAcc[8];
#pragma unroll
  for (int c = 0; c < 8; ++c)
#pragma unroll
    for (int i = 0; i < 8; ++i) ctxAcc[c][i] = 0.0f;
  float rowM = -1e30f, rowL = 0.0f;   // live in lanes 0..15 (one row each)

  const int ntiles = (qBlock + QROWS) / KV;   // causal upper bound for block

  // ---- prologue: kick off tile 0 into buffer 0
  if (wave == 0) {
    tdm_load_tile_128((unsigned)(size_t)&Ks[0][0][0], Kp);
    tdm_load_tile_128((unsigned)(size_t)&Vs[0][0][0], Vp);
  }

  for (int tkv = 0; tkv < ntiles; ++tkv) {
    const int kvBase = tkv * KV;
    const int cur = tkv & 1;
    if (wave == 0) {
      if (tkv + 1 < ntiles) {
        const int nxt = (tkv + 1) & 1;
        tdm_load_tile_128((unsigned)(size_t)&Ks[nxt][0][0],
                          Kp + (size_t)(kvBase + KV) * HN);
        tdm_load_tile_128((unsigned)(size_t)&Vs[nxt][0][0],
                          Vp + (size_t)(kvBase + KV) * HN);
        __builtin_amdgcn_s_wait_tensorcnt(2);   // tile t landed, t+1 in flight
      } else {
        __builtin_amdgcn_s_wait_tensorcnt(0);
      }
    }
    __syncthreads();   // publish buffer `cur` to all waves

    // ---- S = Q * K^T (16x16 per wave), K-dim = HN
    v8f sacc = {};
#pragma unroll 4
    for (int h0 = 0; h0 < HN; h0 += 4) {
      const int hc = h0 + 2 * grp;
      v2f a, b;
      a.x = Qs[wave * 16 + ln16][hc];
      a.y = Qs[wave * 16 + ln16][hc + 1];
      b.x = Ks[cur][ln16][hc];
      b.y = Ks[cur][ln16][hc + 1];
      sacc = wmma_f32(a, b, sacc);
    }
#pragma unroll
    for (int i = 0; i < 8; ++i)
      Ssc[wave][i + 8 * grp][ln16] = sacc[i] * ATT_SCALE;
    __syncthreads();

    // ---- online softmax: lanes 0..15 each own one query row
    if (grp == 0) {
      const int r = ln16;
      const int sAbs = qWave + r;
      float sc[16];
      float tmax = -1e30f;
#pragma unroll
      for (int j = 0; j < 16; ++j) {
        float sv = Ssc[wave][r][j];
        if (kvBase + j > sAbs) sv = MASK_FILL;
        sc[j] = sv;
        tmax = fmaxf(tmax, sv);
      }
      float mNew  = fmaxf(rowM, tmax);
      float alpha = __expf(rowM - mNew);
      float psum  = 0.0f;
#pragma unroll
      for (int j = 0; j < 16; ++j) {
        float p = __expf(sc[j] - mNew);
        Ssc[wave][r][j] = p;
        psum += p;
      }
      rowL = rowL * alpha + psum;
      rowM = mNew;
      aLds[wave][r] = alpha;
    }
    __syncthreads();

    // ---- rescale running ctx and accumulate P * V
#pragma unroll
    for (int c = 0; c < 8; ++c) {
#pragma unroll
      for (int i = 0; i < 8; ++i) ctxAcc[c][i] *= aLds[wave][i + 8 * grp];
    }
#pragma unroll
    for (int c = 0; c < 8; ++c) {
#pragma unroll
      for (int k0 = 0; k0 < 16; k0 += 4) {
        v2f a, b;
        a.x = Ssc[wave][ln16][k0 + 2 * grp];
        a.y = Ssc[wave][ln16][k0 + 2 * grp + 1];
        b.x = Vs[cur][k0 + 2 * grp][c * 16 + ln16];
        b.y = Vs[cur][k0 + 2 * grp + 1][c * 16 + ln16];
        ctxAcc[c] = wmma_f32(a, b, ctxAcc[c]);
      }
    }
    __syncthreads();   // all waves done with buffer `cur` before it is refilled
  }

  // ---- finalize: divide by row sums, store ctx as (s, b, h)
  if (grp == 0) lLds[wave][ln16] = 1.0f / rowL;
  __syncthreads();
  const int bidx = blockIdx.z, nh = blockIdx.y;
#pragma unroll
  for (int c = 0; c < 8; ++c) {
#pragma unroll
    for (int i = 0; i < 8; ++i) {
      int m = i + 8 * grp;
      int sAbs = qWave + m;
      float val = ctxAcc[c][i] * lLds[wave][m];
      size_t dst = ((size_t)sAbs * BATCH + bidx) * HID + (size_t)nh * HN + c * 16 + ln16;
      ctx[dst] = val;
    }
  }
}

// ================================================================ kernel 4
// out = ctx(M_ROWS x HID) * w_dense(HID x HID)^T + b_dense
__global__ __launch_bounds__(256)
void dense_gemm_kernel(const float* __restrict__ A,
                       const float* __restrict__ W,
                       const float* __restrict__ bias,
                       float* __restrict__ out)
{
  __shared__ float As[BM][KPAD];
  __shared__ float Bs[BN][KPAD];

  const int rowBase = blockIdx.y * BM;
  const int colBase = blockIdx.x * BN;
  const int t    = threadIdx.x;
  const int wave = t >> 5;
  const int lane = t & 31;
  const int wm   = wave >> 2;
  const int wn   = wave & 3;
  const int grp  = lane >> 4;
  const int ln16 = lane & 15;

  v8f acc[2][2];
#pragma unroll
  for (int sm = 0; sm < 2; ++sm)
#pragma unroll
    for (int sn = 0; sn < 2; ++sn)
#pragma unroll
      for (int i = 0; i < 8; ++i) acc[sm][sn][i] = 0.0f;

  for (int k0 = 0; k0 < HID; k0 += KT) {
    __syncthreads();
    for (int idx = t; idx < BM * KT; idx += 256) {
      int r = idx >> 5, c = idx & 31;
      As[r][c] = A[(size_t)(rowBase + r) * HID + k0 + c];
      __builtin_prefetch(&A[(size_t)(rowBase + r) * HID + k0 + KT + c], 0, 0);
    }
    for (int idx = t; idx < BN * KT; idx += 256) {
      int r = idx >> 5, c = idx & 31;
      Bs[r][c] = W[(size_t)(colBase + r) * HID + k0 + c];
      __builtin_prefetch(&W[(size_t)(colBase + r) * HID + k0 + KT + c], 0, 0);
    }
    __syncthreads();

#pragma unroll
    for (int kk = 0; kk < KT; kk += 4) {
      const int kc = kk + 2 * grp;
      v2f a0, a1, b0, b1;
      a0.x = As[wm * 32 + ln16][kc];      a0.y = As[wm * 32 + ln16][kc + 1];
      a1.x = As[wm * 32 + 16 + ln16][kc]; a1.y = As[wm * 32 + 16 + ln16][kc + 1];
      b0.x = Bs[wn * 32 + ln16][kc];      b0.y = Bs[wn * 32 + ln16][kc + 1];
      b1.x = Bs[wn * 32 + 16 + ln16][kc]; b1.y = Bs[wn * 32 + 16 + ln16][kc + 1];
      acc[0][0] = wmma_f32(a0, b0, acc[0][0]);
      acc[0][1] = wmma_f32(a0, b1, acc[0][1]);
      acc[1][0] = wmma_f32(a1, b0, acc[1][0]);
      acc[1][1] = wmma_f32(a1, b1, acc[1][1]);
    }
  }

#pragma unroll
  for (int sm = 0; sm < 2; ++sm) {
#pragma unroll
    for (int sn = 0; sn < 2; ++sn) {
#pragma unroll
      for (int i = 0; i < 8; ++i) {
        int m = i + 8 * grp;
        int r = rowBase + wm * 32 + sm * 16 + m;
        int o = colBase + wn * 32 + sn * 16 + ln16;
        out[(size_t)r * HID + o] = acc[sm][sn][i] + bias[o];
      }
    }
  }
}

// ================================================================ launch
extern "C" void kernel_launch(void* const* d_in, const int* in_sizes, int n_in,
                              void* d_out, int out_size, void* d_ws, size_t ws_size,
                              hipStream_t stream)
{
  const float* hidden  = (const float*)d_in[0];
  // d_in[1] = attention_mask (causal; recomputed analytically, not read)
  const float* w_qkv   = (const float*)d_in[2];
  const float* b_qkv   = (const float*)d_in[3];
  const float* w_dense = (const float*)d_in[4];
  const float* b_dense = (const float*)d_in[5];
  float* out = (float*)d_out;

  const size_t headElems = (size_t)BATCH * NHEAD * S_LEN * HN;  // 8M floats
  float* qws = (float*)d_ws;
  float* kws = qws + headElems;
  float* vws = kws + headElems;
  float* ctx = vws + headElems;

  qkv_gemm_kernel<<<dim3(QKV_N / BN, M_ROWS / BM), 256, 0, stream>>>(
      hidden, w_qkv, b_qkv, qws, kws, vws);

  const size_t pairs = (size_t)BATCH * NHEAD * S_LEN * (HN / 2);
  rope_kernel<<<dim3((unsigned)((pairs + 255) / 256)), 256, 0, stream>>>(qws, kws);

  attn_kernel<<<dim3(S_LEN / QROWS, NHEAD, BATCH), 256, 0, stream>>>(
      qws, kws, vws, ctx);

  dense_gemm_kernel<<<dim3(HID / BN, M_ROWS / BM), 256, 0, stream>>>(
      ctx, w_dense, b_dense, out);
}